// GCN_39307540693880
// MI455X (gfx1250) — compile-verified
//
#include <hip/hip_runtime.h>
#include <math.h>

#define GN      2048
#define GNFEAT  128
#define GNHID   32
#define GNCLASS 16
#define KSPLIT  8
#define KCHUNK  (GN / KSPLIT)   // 256
#define MTILES  (GN / 16)       // 128

typedef __attribute__((ext_vector_type(2))) float v2f;
typedef __attribute__((ext_vector_type(8))) float v8f;

__device__ __forceinline__ v8f wmma4(v2f a, v2f b, v8f c) {
  // D = A(16x4 f32) * B(4x16 f32) + C(16x16 f32)
  return __builtin_amdgcn_wmma_f32_16x16x4_f32(false, a, false, b, (short)0, c, false, false);
}

// ---------- Stage A: xW1 = x_init @ W1   [2048,32] ----------
__global__ void k_xW1(const float* __restrict__ x, const float* __restrict__ W1,
                      float* __restrict__ out) {
  int idx = blockIdx.x * blockDim.x + threadIdx.x;   // 65536
  int row = idx >> 5, col = idx & 31;
  const float* xr = x + row * GNFEAT;
  float s = 0.f;
#pragma unroll 8
  for (int k = 0; k < GNFEAT; ++k) s = fmaf(xr[k], W1[k * GNHID + col], s);
  out[idx] = s;
}

// ---------- Stage B: partB[ks][mt] = adj[16 rows, Kchunk] @ xW1[Kchunk, 32] ----------
// one wave per (M-tile, K-chunk); two 16x16 N-halves per wave so adj is read once
__global__ void k_gemmB(const float* __restrict__ adj, const float* __restrict__ B,
                        float* __restrict__ part) {
  const int mt = blockIdx.x, ks = blockIdx.y;
  const int lane = threadIdx.x;          // 0..31, EXEC all ones
  const int m = lane & 15, khi = lane >> 4;
  const int row = mt * 16 + m;
  const float* arow = adj + (size_t)row * GN;
  const int kbeg = ks * KCHUNK, kend = kbeg + KCHUNK;
  v8f acc0a = {}, acc0b = {}, acc1a = {}, acc1b = {};
  for (int k = kbeg; k < kend; k += 8) {
    int kk0 = k + 2 * khi;
    int kk1 = k + 4 + 2 * khi;
    v2f a0 = *(const v2f*)(arow + kk0);                           // A: M=m, K=kk0..kk0+1
    v2f a1 = *(const v2f*)(arow + kk1);
    v2f b00 = { B[kk0 * GNHID + m],      B[(kk0 + 1) * GNHID + m]      };
    v2f b01 = { B[kk0 * GNHID + m + 16], B[(kk0 + 1) * GNHID + m + 16] };
    v2f b10 = { B[kk1 * GNHID + m],      B[(kk1 + 1) * GNHID + m]      };
    v2f b11 = { B[kk1 * GNHID + m + 16], B[(kk1 + 1) * GNHID + m + 16] };
    acc0a = wmma4(a0, b00, acc0a);
    acc1a = wmma4(a0, b01, acc1a);
    acc0b = wmma4(a1, b10, acc0b);
    acc1b = wmma4(a1, b11, acc1b);
  }
  v8f acc0 = acc0a + acc0b, acc1 = acc1a + acc1b;
  float* pt = part + (size_t)(ks * MTILES + mt) * (16 * 32);
#pragma unroll
  for (int r = 0; r < 8; ++r) {
    int Md = r + 8 * khi;                 // D: lanes 0-15 -> M=r, lanes 16-31 -> M=r+8
    pt[Md * 32 + m]      = acc0[r];
    pt[Md * 32 + m + 16] = acc1[r];
  }
}

// ---------- reduce partials + bias + tanh -> H ----------
__global__ void k_reduceB(const float* __restrict__ part, const float* __restrict__ b1,
                          float* __restrict__ H) {
  int idx = blockIdx.x * blockDim.x + threadIdx.x;   // 65536
  int row = idx >> 5, col = idx & 31;
  int mt = row >> 4, m = row & 15;
  float s = b1[col];
#pragma unroll
  for (int ks = 0; ks < KSPLIT; ++ks)
    s += part[(size_t)(ks * MTILES + mt) * 512 + m * 32 + col];
  H[idx] = tanhf(s);
}

// ---------- fold confidence MLP: wa = Wa@w, wb = Wb@w, c0 = bc1.w + bc2 ----------
__global__ void k_conf(const float* __restrict__ Wc1, const float* __restrict__ bc1,
                       const float* __restrict__ Wc2, const float* __restrict__ bc2,
                       float* __restrict__ wab) {   // wab[0:32]=wa, [32:64]=wb, [64]=c0
  int t = threadIdx.x;   // 64 threads, t indexes rows of Wc1
  float s = 0.f;
#pragma unroll
  for (int h = 0; h < GNHID; ++h) s = fmaf(Wc1[t * GNHID + h], Wc2[h], s);
  wab[t] = s;
  if (t == 0) {
    float c = bc2[0];
    for (int h = 0; h < GNHID; ++h) c = fmaf(bc1[h], Wc2[h], c);
    wab[64] = c;
  }
}

// ---------- u[i] = x_i.wa + c0 ; v[i] = x_i.wb ----------
__global__ void k_uv(const float* __restrict__ H, const float* __restrict__ wab,
                     float* __restrict__ u, float* __restrict__ v) {
  int row = blockIdx.x * blockDim.x + threadIdx.x;   // 2048
  const float* h = H + row * GNHID;
  float su = wab[64], sv = 0.f;
#pragma unroll
  for (int q = 0; q < GNHID; ++q) {
    float x = h[q];
    su = fmaf(x, wab[q], su);
    sv = fmaf(x, wab[32 + q], sv);
  }
  u[row] = su; v[row] = sv;
}

// ---------- xcW2 = [H, x_init] @ W2   [2048,16] ----------
__global__ void k_xcW2(const float* __restrict__ H, const float* __restrict__ x,
                       const float* __restrict__ W2, float* __restrict__ out) {
  int idx = blockIdx.x * blockDim.x + threadIdx.x;   // 32768
  int row = idx >> 4, col = idx & 15;
  float s = 0.f;
  const float* h = H + row * GNHID;
#pragma unroll 8
  for (int q = 0; q < GNHID; ++q) s = fmaf(h[q], W2[q * GNCLASS + col], s);
  const float* xr = x + row * GNFEAT;
#pragma unroll 8
  for (int k = 0; k < GNFEAT; ++k) s = fmaf(xr[k], W2[(GNHID + k) * GNCLASS + col], s);
  out[idx] = s;
}

// ---------- Stage E: partE[ks][mt] = (adj * score(u_i + v_j)) @ xcW2 ----------
__global__ void k_gemmE(const float* __restrict__ adj, const float* __restrict__ B,
                        const float* __restrict__ u, const float* __restrict__ v,
                        float* __restrict__ part) {
  const int mt = blockIdx.x, ks = blockIdx.y;
  const int lane = threadIdx.x;
  const int m = lane & 15, khi = lane >> 4;
  const int row = mt * 16 + m;
  const float ui = u[row];
  const float* arow = adj + (size_t)row * GN;
  const int kbeg = ks * KCHUNK, kend = kbeg + KCHUNK;
  v8f acca = {}, accb = {};
  for (int k = kbeg; k < kend; k += 8) {
    int kk0 = k + 2 * khi;
    int kk1 = k + 4 + 2 * khi;
    v2f a0 = *(const v2f*)(arow + kk0);
    v2f a1 = *(const v2f*)(arow + kk1);
    v2f v0 = *(const v2f*)(v + kk0);
    v2f v1 = *(const v2f*)(v + kk1);
    float e;
    e = ui + v0.x; a0.x *= (e > 0.f) ? 1.f : e;   // masked_fill(e>0, 1)
    e = ui + v0.y; a0.y *= (e > 0.f) ? 1.f : e;
    e = ui + v1.x; a1.x *= (e > 0.f) ? 1.f : e;
    e = ui + v1.y; a1.y *= (e > 0.f) ? 1.f : e;
    v2f b0 = { B[kk0 * GNCLASS + m], B[(kk0 + 1) * GNCLASS + m] };
    v2f b1 = { B[kk1 * GNCLASS + m], B[(kk1 + 1) * GNCLASS + m] };
    acca = wmma4(a0, b0, acca);
    accb = wmma4(a1, b1, accb);
  }
  v8f acc = acca + accb;
  float* pt = part + (size_t)(ks * MTILES + mt) * (16 * 16);
#pragma unroll
  for (int r = 0; r < 8; ++r) {
    int Md = r + 8 * khi;
    pt[Md * 16 + m] = acc[r];
  }
}

// ---------- reduce partials + bias -> pre-softmax logits ----------
__global__ void k_reduceE(const float* __restrict__ part, const float* __restrict__ b2,
                          float* __restrict__ out) {
  int idx = blockIdx.x * blockDim.x + threadIdx.x;   // 32768
  int row = idx >> 4, col = idx & 15;
  int mt = row >> 4, m = row & 15;
  float s = b2[col];
#pragma unroll
  for (int ks = 0; ks < KSPLIT; ++ks)
    s += part[(size_t)(ks * MTILES + mt) * 256 + m * 16 + col];
  out[idx] = s;
}

// ---------- row-wise log_softmax over 16 classes ----------
__global__ void k_lsm(const float* __restrict__ pre, float* __restrict__ out) {
  int row = blockIdx.x * blockDim.x + threadIdx.x;   // 2048
  const float* p = pre + row * GNCLASS;
  float mx = p[0];
#pragma unroll
  for (int c = 1; c < GNCLASS; ++c) mx = fmaxf(mx, p[c]);
  float se = 0.f;
#pragma unroll
  for (int c = 0; c < GNCLASS; ++c) se += expf(p[c] - mx);
  float lse = logf(se) + mx;
#pragma unroll
  for (int c = 0; c < GNCLASS; ++c) out[row * GNCLASS + c] = p[c] - lse;
}

extern "C" void kernel_launch(void* const* d_in, const int* in_sizes, int n_in,
                              void* d_out, int out_size, void* d_ws, size_t ws_size,
                              hipStream_t stream) {
  (void)in_sizes; (void)n_in; (void)out_size; (void)ws_size;
  const float* x_init = (const float*)d_in[0];
  const float* adj1   = (const float*)d_in[2];   // adj1_org; d_in[1]/d_in[3] unused by reference
  const float* W1  = (const float*)d_in[4];
  const float* b1  = (const float*)d_in[5];
  const float* W2  = (const float*)d_in[6];
  const float* b2  = (const float*)d_in[7];
  const float* Wc1 = (const float*)d_in[8];
  const float* bc1 = (const float*)d_in[9];
  const float* Wc2 = (const float*)d_in[10];
  const float* bc2 = (const float*)d_in[11];
  float* out = (float*)d_out;

  float* ws    = (float*)d_ws;
  float* xW1   = ws;                    // 65536
  float* H     = xW1   + 65536;         // 65536
  float* partB = H     + 65536;         // 524288
  float* partE = partB + 524288;        // 262144
  float* wab   = partE + 262144;        // 65 (padded to 96)
  float* u     = wab   + 96;            // 2048
  float* v     = u     + 2048;          // 2048
  float* xcW2  = v     + 2048;          // 32768
  float* pre   = xcW2  + 32768;         // 32768  (total ~3.95 MB)

  k_xW1    <<<256, 256, 0, stream>>>(x_init, W1, xW1);
  k_gemmB  <<<dim3(MTILES, KSPLIT), 32, 0, stream>>>(adj1, xW1, partB);
  k_reduceB<<<256, 256, 0, stream>>>(partB, b1, H);
  k_conf   <<<1, 64, 0, stream>>>(Wc1, bc1, Wc2, bc2, wab);
  k_uv     <<<8, 256, 0, stream>>>(H, wab, u, v);
  k_xcW2   <<<128, 256, 0, stream>>>(H, x_init, W2, xcW2);
  k_gemmE  <<<dim3(MTILES, KSPLIT), 32, 0, stream>>>(adj1, xcW2, u, v, partE);
  k_reduceE<<<128, 256, 0, stream>>>(partE, b2, pre);
  k_lsm    <<<8, 256, 0, stream>>>(pre, out);
}